// Model_83519934038715
// MI455X (gfx1250) — compile-verified
//
#include <hip/hip_runtime.h>
#include <math.h>

// Problem constants (match reference)
#define NN   50000
#define EE   800000
#define INF_ 128
#define HH   64
#define OUTD 64
#define TOTAL_STEPS 55          // 50 max while-loop iters + 5 phantom steps
#define NT   (NN / 16)          // 3125 WMMA row tiles (N divisible by 16)
#define WPB  4                  // waves per block for GEMM kernels

typedef float v2f __attribute__((ext_vector_type(2)));
typedef float v8f __attribute__((ext_vector_type(8)));

__device__ __forceinline__ float sigmoidf_(float v) {
    return 1.0f / (1.0f + __expf(-v));
}

// ---------------------------------------------------------------- utilities
__global__ void zero_i32_kernel(int* __restrict__ p, int n) {
    int i = blockIdx.x * blockDim.x + threadIdx.x;
    if (i < n) p[i] = 0;
}

// ---------------------------------------------------------------- CSR build
__global__ void count_deg_kernel(const int* __restrict__ dst,
                                 int* __restrict__ counts, int e) {
    int i = blockIdx.x * blockDim.x + threadIdx.x;
    if (i < e) atomicAdd(&counts[dst[i]], 1);
}

// single-workgroup Hillis-Steele scan over chunks of 1024
__global__ __launch_bounds__(1024)
void scan_rowptr_kernel(const int* __restrict__ counts,
                        int* __restrict__ row_ptr, int n) {
    __shared__ int sdata[1024];
    __shared__ int s_running;
    int tid = threadIdx.x;
    if (tid == 0) { s_running = 0; row_ptr[0] = 0; }
    __syncthreads();
    for (int base = 0; base < n; base += 1024) {
        int i = base + tid;
        int v = (i < n) ? counts[i] : 0;
        sdata[tid] = v;
        __syncthreads();
        for (int off = 1; off < 1024; off <<= 1) {
            int t = (tid >= off) ? sdata[tid - off] : 0;
            __syncthreads();
            if (tid >= off) sdata[tid] += t;
            __syncthreads();
        }
        int incl = sdata[tid];
        int run  = s_running;
        if (i < n) row_ptr[i + 1] = run + incl;
        __syncthreads();
        if (tid == 0) s_running = run + sdata[1023];
        __syncthreads();
    }
}

__global__ void fill_csr_kernel(const int* __restrict__ src,
                                const int* __restrict__ dst,
                                const float* __restrict__ ew,
                                const int* __restrict__ row_ptr,
                                int* __restrict__ cursor,
                                int* __restrict__ csr_src,
                                float* __restrict__ csr_w, int e) {
    int i = blockIdx.x * blockDim.x + threadIdx.x;
    if (i >= e) return;
    int d = dst[i];
    int pos = row_ptr[d] + atomicAdd(&cursor[d], 1);
    csr_src[pos] = src[i];
    csr_w[pos]   = ew[i];
}

// ------------------------------------------------- encoder + bias (2 GEMMs)
// b = (x @ enc_w^T) @ bias_w^T + bias_b, one wave per 16x64 output tile.
__global__ __launch_bounds__(WPB * 32)
void enc_bias_kernel(const float* __restrict__ x,
                     const float* __restrict__ enc_w,
                     const float* __restrict__ bias_w,
                     const float* __restrict__ bias_b,
                     float* __restrict__ bout) {
    __shared__ float hstage[WPB][16][HH + 4];   // pad to avoid bank conflicts
    const int w     = threadIdx.x >> 5;
    const int lane  = threadIdx.x & 31;
    const int tile  = blockIdx.x * WPB + w;
    if (tile >= NT) return;
    const int row0  = tile * 16;
    const int halfl = lane & 15;
    const int hi    = lane >> 4;                 // 0: K+0/1, 1: K+2/3

    // ---- GEMM1: h_tile[16x64] = x_tile[16x128] @ enc_w^T[128x64]
    v8f c0 = {}, c1 = {}, c2 = {}, c3 = {};
    const int arow = row0 + halfl;
    for (int k = 0; k < INF_; k += 4) {
        const int kk = k + 2 * hi;
        v2f a;
        a.x = x[arow * INF_ + kk];
        a.y = x[arow * INF_ + kk + 1];
        v2f bm;
        int col;
        col = 0 * 16 + halfl;
        bm.x = enc_w[col * INF_ + kk]; bm.y = enc_w[col * INF_ + kk + 1];
        c0 = __builtin_amdgcn_wmma_f32_16x16x4_f32(false, a, false, bm, (short)0, c0, false, false);
        col = 1 * 16 + halfl;
        bm.x = enc_w[col * INF_ + kk]; bm.y = enc_w[col * INF_ + kk + 1];
        c1 = __builtin_amdgcn_wmma_f32_16x16x4_f32(false, a, false, bm, (short)0, c1, false, false);
        col = 2 * 16 + halfl;
        bm.x = enc_w[col * INF_ + kk]; bm.y = enc_w[col * INF_ + kk + 1];
        c2 = __builtin_amdgcn_wmma_f32_16x16x4_f32(false, a, false, bm, (short)0, c2, false, false);
        col = 3 * 16 + halfl;
        bm.x = enc_w[col * INF_ + kk]; bm.y = enc_w[col * INF_ + kk + 1];
        c3 = __builtin_amdgcn_wmma_f32_16x16x4_f32(false, a, false, bm, (short)0, c3, false, false);
    }

    // ---- stage h tile to LDS (C-layout -> row-major), wave-private region
#pragma unroll
    for (int j = 0; j < 8; ++j) {
        const int m = j + 8 * hi;
        hstage[w][m][0 * 16 + halfl] = c0[j];
        hstage[w][m][1 * 16 + halfl] = c1[j];
        hstage[w][m][2 * 16 + halfl] = c2[j];
        hstage[w][m][3 * 16 + halfl] = c3[j];
    }

    // ---- GEMM2: b_tile = h_tile[16x64] @ bias_w^T[64x64] + bias_b
    v8f d0, d1, d2, d3;
    {
        const float bb0 = bias_b[0 * 16 + halfl];
        const float bb1 = bias_b[1 * 16 + halfl];
        const float bb2 = bias_b[2 * 16 + halfl];
        const float bb3 = bias_b[3 * 16 + halfl];
#pragma unroll
        for (int j = 0; j < 8; ++j) { d0[j] = bb0; d1[j] = bb1; d2[j] = bb2; d3[j] = bb3; }
    }
    for (int k = 0; k < HH; k += 4) {
        const int kk = k + 2 * hi;
        v2f a;
        a.x = hstage[w][halfl][kk];
        a.y = hstage[w][halfl][kk + 1];
        v2f bm;
        int col;
        col = 0 * 16 + halfl;
        bm.x = bias_w[col * HH + kk]; bm.y = bias_w[col * HH + kk + 1];
        d0 = __builtin_amdgcn_wmma_f32_16x16x4_f32(false, a, false, bm, (short)0, d0, false, false);
        col = 1 * 16 + halfl;
        bm.x = bias_w[col * HH + kk]; bm.y = bias_w[col * HH + kk + 1];
        d1 = __builtin_amdgcn_wmma_f32_16x16x4_f32(false, a, false, bm, (short)0, d1, false, false);
        col = 2 * 16 + halfl;
        bm.x = bias_w[col * HH + kk]; bm.y = bias_w[col * HH + kk + 1];
        d2 = __builtin_amdgcn_wmma_f32_16x16x4_f32(false, a, false, bm, (short)0, d2, false, false);
        col = 3 * 16 + halfl;
        bm.x = bias_w[col * HH + kk]; bm.y = bias_w[col * HH + kk + 1];
        d3 = __builtin_amdgcn_wmma_f32_16x16x4_f32(false, a, false, bm, (short)0, d3, false, false);
    }

#pragma unroll
    for (int j = 0; j < 8; ++j) {
        const int m = row0 + j + 8 * hi;
        bout[m * HH + 0 * 16 + halfl] = d0[j];
        bout[m * HH + 1 * 16 + halfl] = d1[j];
        bout[m * HH + 2 * 16 + halfl] = d2[j];
        bout[m * HH + 3 * 16 + halfl] = d3[j];
    }
}

// --------------------------------------------------- z0 step: z = relu(bs*b)
__global__ void init_z_kernel(const float* __restrict__ b,
                              const float* __restrict__ beta,
                              float* __restrict__ z) {
    int i = blockIdx.x * blockDim.x + threadIdx.x;
    const float bs = sigmoidf_(beta[0]);
    if (i < NN * HH / 4) {
        const float4 bv = ((const float4*)b)[i];
        float4 o;
        o.x = fmaxf(0.f, bs * bv.x);
        o.y = fmaxf(0.f, bs * bv.y);
        o.z = fmaxf(0.f, bs * bv.z);
        o.w = fmaxf(0.f, bs * bv.w);
        ((float4*)z)[i] = o;
    }
}

// ------------------------------------- fused SpMM + affine + ReLU iteration
// one wave per destination row; lane handles 2 of the 64 features
__global__ __launch_bounds__(256)
void spmm_step_kernel(const int* __restrict__ row_ptr,
                      const int* __restrict__ csr_src,
                      const float* __restrict__ csr_w,
                      const float* __restrict__ z_in,
                      const float* __restrict__ b,
                      const float* __restrict__ gamma,
                      const float* __restrict__ beta,
                      float* __restrict__ z_out) {
    const int wave = (blockIdx.x * blockDim.x + threadIdx.x) >> 5;
    const int lane = threadIdx.x & 31;
    if (wave >= NN) return;
    const int row   = wave;
    const int begin = row_ptr[row];
    const int end   = row_ptr[row + 1];
    const int c     = lane * 2;
    float2 acc = make_float2(0.f, 0.f);
    for (int p = begin; p < end; ++p) {
        const int   s = csr_src[p];      // uniform across the wave -> SGPR
        const float wgt = csr_w[p];
        const float2 zv = *(const float2*)(z_in + s * HH + c);
        acc.x += wgt * zv.x;
        acc.y += wgt * zv.y;
    }
    const float gs = sigmoidf_(gamma[0]);
    const float bs = sigmoidf_(beta[0]);
    const float2 bv = *(const float2*)(b + row * HH + c);
    float2 o;
    o.x = fmaxf(0.f, gs * acc.x + bs * bv.x);
    o.y = fmaxf(0.f, gs * acc.y + bs * bv.y);
    *(float2*)(z_out + row * HH + c) = o;
}

// ----------------------------------------------------- decoder GEMM (WMMA)
// out = relu(z) @ dec_w^T + dec_b, one wave per 16x64 tile
__global__ __launch_bounds__(WPB * 32)
void decoder_kernel(const float* __restrict__ z,
                    const float* __restrict__ dec_w,
                    const float* __restrict__ dec_b,
                    float* __restrict__ out) {
    const int w     = threadIdx.x >> 5;
    const int lane  = threadIdx.x & 31;
    const int tile  = blockIdx.x * WPB + w;
    if (tile >= NT) return;
    const int row0  = tile * 16;
    const int halfl = lane & 15;
    const int hi    = lane >> 4;

    v8f d0, d1, d2, d3;
    {
        const float bb0 = dec_b[0 * 16 + halfl];
        const float bb1 = dec_b[1 * 16 + halfl];
        const float bb2 = dec_b[2 * 16 + halfl];
        const float bb3 = dec_b[3 * 16 + halfl];
#pragma unroll
        for (int j = 0; j < 8; ++j) { d0[j] = bb0; d1[j] = bb1; d2[j] = bb2; d3[j] = bb3; }
    }
    const int arow = row0 + halfl;
    for (int k = 0; k < HH; k += 4) {
        const int kk = k + 2 * hi;
        v2f a;
        a.x = fmaxf(0.f, z[arow * HH + kk]);       // relu (idempotent)
        a.y = fmaxf(0.f, z[arow * HH + kk + 1]);
        v2f bm;
        int col;
        col = 0 * 16 + halfl;
        bm.x = dec_w[col * HH + kk]; bm.y = dec_w[col * HH + kk + 1];
        d0 = __builtin_amdgcn_wmma_f32_16x16x4_f32(false, a, false, bm, (short)0, d0, false, false);
        col = 1 * 16 + halfl;
        bm.x = dec_w[col * HH + kk]; bm.y = dec_w[col * HH + kk + 1];
        d1 = __builtin_amdgcn_wmma_f32_16x16x4_f32(false, a, false, bm, (short)0, d1, false, false);
        col = 2 * 16 + halfl;
        bm.x = dec_w[col * HH + kk]; bm.y = dec_w[col * HH + kk + 1];
        d2 = __builtin_amdgcn_wmma_f32_16x16x4_f32(false, a, false, bm, (short)0, d2, false, false);
        col = 3 * 16 + halfl;
        bm.x = dec_w[col * HH + kk]; bm.y = dec_w[col * HH + kk + 1];
        d3 = __builtin_amdgcn_wmma_f32_16x16x4_f32(false, a, false, bm, (short)0, d3, false, false);
    }
#pragma unroll
    for (int j = 0; j < 8; ++j) {
        const int m = row0 + j + 8 * hi;
        out[m * OUTD + 0 * 16 + halfl] = d0[j];
        out[m * OUTD + 1 * 16 + halfl] = d1[j];
        out[m * OUTD + 2 * 16 + halfl] = d2[j];
        out[m * OUTD + 3 * 16 + halfl] = d3[j];
    }
}

// ------------------------------------------------------------------- driver
extern "C" void kernel_launch(void* const* d_in, const int* in_sizes, int n_in,
                              void* d_out, int out_size, void* d_ws, size_t ws_size,
                              hipStream_t stream) {
    const float* x      = (const float*)d_in[0];
    const int*   ei     = (const int*)d_in[1];     // [2,E]: src then dst
    const float* ew     = (const float*)d_in[2];
    const float* enc_w  = (const float*)d_in[3];
    const float* bias_w = (const float*)d_in[4];
    const float* bias_b = (const float*)d_in[5];
    const float* dec_w  = (const float*)d_in[6];
    const float* dec_b  = (const float*)d_in[7];
    const float* beta   = (const float*)d_in[8];
    const float* gamma  = (const float*)d_in[9];
    float* out = (float*)d_out;

    char* ws = (char*)d_ws;
    const size_t ZBYTES = (size_t)NN * HH * sizeof(float);    // 12.8 MB
    float* b_buf   = (float*)(ws);
    float* zA      = (float*)(ws + ZBYTES);
    float* zB      = (float*)(ws + 2 * ZBYTES);
    int*   row_ptr = (int*)  (ws + 3 * ZBYTES);
    int*   counts  = (int*)  (ws + 3 * ZBYTES + 256 * (((NN + 1) * 4 + 255) / 256));
    int*   csr_src = (int*)  (ws + 3 * ZBYTES + 2 * 256 * (((NN + 1) * 4 + 255) / 256));
    float* csr_w   = (float*)((char*)csr_src + (size_t)EE * 4);

    const int* src = ei;
    const int* dst = ei + EE;

    // ---- CSR build (amortized over the 55-step fixed-point loop)
    zero_i32_kernel<<<(NN + 255) / 256, 256, 0, stream>>>(counts, NN);
    count_deg_kernel<<<(EE + 255) / 256, 256, 0, stream>>>(dst, counts, EE);
    scan_rowptr_kernel<<<1, 1024, 0, stream>>>(counts, row_ptr, NN);
    zero_i32_kernel<<<(NN + 255) / 256, 256, 0, stream>>>(counts, NN);
    fill_csr_kernel<<<(EE + 255) / 256, 256, 0, stream>>>(src, dst, ew, row_ptr,
                                                          counts, csr_src, csr_w, EE);

    // ---- b = (x @ enc_w^T) @ bias_w^T + bias_b   (fp32 WMMA, exact)
    enc_bias_kernel<<<(NT + WPB - 1) / WPB, WPB * 32, 0, stream>>>(
        x, enc_w, bias_w, bias_b, b_buf);

    // ---- fixed-point: z1 = relu(bs*b); then 54 more fused SpMM steps (55 total)
    init_z_kernel<<<(NN * HH / 4 + 255) / 256, 256, 0, stream>>>(b_buf, beta, zA);
    float* zin = zA;
    float* zout = zB;
    for (int it = 0; it < TOTAL_STEPS - 1; ++it) {
        spmm_step_kernel<<<(NN * 32 + 255) / 256, 256, 0, stream>>>(
            row_ptr, csr_src, csr_w, zin, b_buf, gamma, beta, zout);
        float* t = zin; zin = zout; zout = t;
    }

    // ---- decoder: out = relu(z) @ dec_w^T + dec_b (fp32 WMMA)
    decoder_kernel<<<(NT + WPB - 1) / WPB, WPB * 32, 0, stream>>>(
        zin, dec_w, dec_b, out);
}